// HistogramLayer_52776558133573
// MI455X (gfx1250) — compile-verified
//
#include <hip/hip_runtime.h>
#include <cmath>

// -----------------------------------------------------------------------------
// HistogramLayer: x (16,10,512,512) f32 ->
//   cos = x[:, :8], grads = x[:, 8:10]
//   mag = sqrt(g8^2 + g9^2); out[:, c] = (argmax_c cos == c) ? mag : 0
// Memory-bound streaming kernel: ~302 MB traffic -> ~13 us roofline @ 23.3 TB/s.
// CDNA5 path: async global->LDS b128 staging (ASYNCcnt) + nontemporal b128 I/O.
// -----------------------------------------------------------------------------

typedef float v4f __attribute__((ext_vector_type(4)));
typedef int   v4i __attribute__((ext_vector_type(4)));

#define AS_GLOBAL __attribute__((address_space(1)))
#define AS_LDS    __attribute__((address_space(3)))

#if defined(__AMDGCN__) && \
    __has_builtin(__builtin_amdgcn_global_load_async_to_lds_b128) && \
    __has_builtin(__builtin_amdgcn_s_wait_asynccnt)
#define USE_ASYNC_LDS 1
#else
#define USE_ASYNC_LDS 0
#endif

static constexpr int kB       = 16;
static constexpr int kH       = 512;
static constexpr int kW       = 512;
static constexpr int kPlane   = kH * kW;     // 262144 floats per channel plane
static constexpr int kCin     = 10;
static constexpr int kCout    = 8;
static constexpr int kThreads = 256;         // 8 wave32 per block

__global__ __launch_bounds__(kThreads)
void histogram_layer_kernel(const float* __restrict__ x, float* __restrict__ out) {
    const int tid = threadIdx.x;
    // Each thread owns 4 consecutive pixels (one float4 per channel).
    const long long quad = (long long)blockIdx.x * kThreads + tid;
    const long long p    = quad * 4;                    // first pixel index
    const int b = (int)(p / kPlane);                    // batch
    const int r = (int)(p - (long long)b * kPlane);     // offset within plane (mult of 4)

    const float* __restrict__ xin = x + (long long)b * kCin * kPlane + r;

    v4f ch[kCin];

#if USE_ASYNC_LDS
    // Stage all 10 channel quads through LDS with the CDNA5 async DMA path.
    // 10 ch * 256 thr * 16 B = 40 KB LDS per block (WGP has 320 KB).
    __shared__ float lds[kCin * kThreads * 4];
    float* myLds = &lds[tid * 4];
    #pragma unroll
    for (int c = 0; c < kCin; ++c) {
        __builtin_amdgcn_global_load_async_to_lds_b128(
            (AS_GLOBAL v4i*)(xin + (long long)c * kPlane),
            (AS_LDS v4i*)(myLds + c * (kThreads * 4)),
            /*imm offset*/ 0, /*cpol*/ 0);
    }
    __builtin_amdgcn_s_wait_asynccnt(0);   // ASYNCcnt == 0: LDS tiles are resident
    #pragma unroll
    for (int c = 0; c < kCin; ++c) {
        ch[c] = *(const v4f*)(myLds + c * (kThreads * 4));  // ds_load_b128, private region
    }
#else
    // Fallback: straight nontemporal b128 loads (data read exactly once).
    #pragma unroll
    for (int c = 0; c < kCin; ++c) {
        ch[c] = __builtin_nontemporal_load((const v4f*)(xin + (long long)c * kPlane));
    }
#endif

    // Per-component argmax over the 8 cosine channels (strict > => first-max wins,
    // matching jnp.argmax tie-breaking) + gradient magnitude.
    int  idx[4];
    v4f  mag;
    #pragma unroll
    for (int k = 0; k < 4; ++k) {
        float bv = ch[0][k];
        int   bi = 0;
        #pragma unroll
        for (int c = 1; c < kCout; ++c) {
            if (ch[c][k] > bv) { bv = ch[c][k]; bi = c; }
        }
        idx[k] = bi;
        const float g0 = ch[8][k];
        const float g1 = ch[9][k];
        mag[k] = __builtin_sqrtf(g0 * g0 + g1 * g1);
    }

    // Scatter one-hot * magnitude into the 8 output channels.
    // Output is written once and never re-read: nontemporal b128 stores keep the
    // 134 MB output stream from evicting the input stream out of L2.
    float* __restrict__ op = out + (long long)b * kCout * kPlane + r;
    #pragma unroll
    for (int c = 0; c < kCout; ++c) {
        v4f o;
        #pragma unroll
        for (int k = 0; k < 4; ++k) {
            o[k] = (idx[k] == c) ? mag[k] : 0.0f;
        }
        __builtin_nontemporal_store(o, (v4f*)(op + (long long)c * kPlane));
    }
}

extern "C" void kernel_launch(void* const* d_in, const int* in_sizes, int n_in,
                              void* d_out, int out_size, void* d_ws, size_t ws_size,
                              hipStream_t stream) {
    (void)in_sizes; (void)n_in; (void)out_size; (void)d_ws; (void)ws_size;

    const float* x   = (const float*)d_in[0];
    float*       out = (float*)d_out;

    // Total pixel-quads: 16 * 262144 / 4 = 1,048,576 -> 4096 blocks of 256.
    const long long quads  = (long long)kB * kPlane / 4;
    const int       blocks = (int)(quads / kThreads);

    histogram_layer_kernel<<<blocks, kThreads, 0, stream>>>(x, out);
}